// GraphClassificationHead_38792144618154
// MI455X (gfx1250) — compile-verified
//
#include <hip/hip_runtime.h>
#include <hip/hip_bf16.h>

// ---------------- problem constants (match reference) ----------------
#define N_NODES   100000
#define N_EDGES   3200000
#define IN_FEATS  256
#define NUM_CLS   16
#define NUM_GRAPHS 256
#define N_TILES   (N_NODES / 16)   // 6250 exactly

typedef float v2f __attribute__((ext_vector_type(2)));
typedef float v8f __attribute__((ext_vector_type(8)));

// ---------------- zero a float region ----------------
__global__ void gch_zero(float* __restrict__ p, long long n) {
    long long i = (long long)blockIdx.x * blockDim.x + threadIdx.x;
    if (i < n) p[i] = 0.0f;
}

// ---------------- degree accumulation (f32 atomics, deg arrays live in L2) ----------------
__global__ void gch_degrees(const long long* __restrict__ src,
                            const long long* __restrict__ dst,
                            float* __restrict__ deg_src,
                            float* __restrict__ deg_dst, int nEdges) {
    int e = blockIdx.x * blockDim.x + threadIdx.x;
    if (e >= nEdges) return;
    atomicAdd(&deg_src[(int)src[e]], 1.0f);
    atomicAdd(&deg_dst[(int)dst[e]], 1.0f);
}

// ---------------- in-place norm = rsqrt(max(deg,1)) ----------------
__global__ void gch_norms(float* __restrict__ a, float* __restrict__ b, int n) {
    int i = blockIdx.x * blockDim.x + threadIdx.x;
    if (i >= n) return;
    a[i] = __frsqrt_rn(fmaxf(a[i], 1.0f));
    b[i] = __frsqrt_rn(fmaxf(b[i], 1.0f));
}

// ---------------- projection: h = (in_feat @ W) * norm_src, via V_WMMA_F32_16X16X4_F32 ----
// One wave computes a 16x16 output tile (16 nodes x 16 classes), looping K=256 in
// 64 steps of the fp32 16x16x4 WMMA. W (256x16 = 16KB) is staged in LDS per block.
// A-fragment layout (ISA 32-bit A 16x4): lanes 0-15 -> K=k,k+1 ; lanes 16-31 -> K=k+2,k+3
// C/D layout: VGPR r, lanes 0-15 -> row r, lanes 16-31 -> row r+8, col = lane&15
__global__ __launch_bounds__(256) void gch_project(const float* __restrict__ A,
                                                   const float* __restrict__ Wm,
                                                   const float* __restrict__ norm_src,
                                                   float* __restrict__ h,
                                                   int nTiles) {
    __shared__ float Ws[IN_FEATS * NUM_CLS];           // 16 KB
    // cooperative W load (float4, fully coalesced)
    for (int i = threadIdx.x; i < IN_FEATS * NUM_CLS / 4; i += 256) {
        ((float4*)Ws)[i] = ((const float4*)Wm)[i];
    }
    __syncthreads();

    const int wave = threadIdx.x >> 5;
    const int lane = threadIdx.x & 31;
    const int tile = blockIdx.x * 8 + wave;
    if (tile >= nTiles) return;                        // wave-uniform: EXEC stays all-1

    const int m    = lane & 15;                        // row within tile / col for B
    const int half = lane >> 4;                        // 0: K=k,k+1  1: K=k+2,k+3
    const long long row = (long long)tile * 16 + m;
    const float* arow = A + row * IN_FEATS + half * 2;

    v8f c = {};
    #pragma unroll 4
    for (int k = 0; k < IN_FEATS; k += 4) {
        v2f a = *(const v2f*)(arow + k);               // global_load_b64, 8B aligned
        const float* wp = Ws + (k + half * 2) * NUM_CLS + m;
        v2f bfrag; bfrag.x = wp[0]; bfrag.y = wp[NUM_CLS];
        c = __builtin_amdgcn_wmma_f32_16x16x4_f32(
                /*neg_a=*/false, a, /*neg_b=*/false, bfrag,
                /*c_mod=*/(short)0, c, /*reuse_a=*/false, /*reuse_b=*/false);
    }

    // scaled store: h[row][col] = c * norm_src[row]; 64B contiguous per half-wave
    const int rowOff = half * 8;
    #pragma unroll
    for (int r = 0; r < 8; ++r) {
        const long long rr = (long long)tile * 16 + r + rowOff;
        h[rr * NUM_CLS + m] = c[r] * norm_src[rr];
    }
}

// ---------------- edge aggregation: agg[dst] += h[src] ----------------
// 16 lanes per edge: lanes of a half-wave gather a contiguous 64B row of h
// (L2-resident), then issue f32 atomics into agg (also L2-resident).
__global__ void gch_aggregate(const long long* __restrict__ src,
                              const long long* __restrict__ dst,
                              const float* __restrict__ h,
                              float* __restrict__ agg, long long nWork) {
    long long gid = (long long)blockIdx.x * blockDim.x + threadIdx.x;
    if (gid >= nWork) return;
    const int e = (int)(gid >> 4);
    const int cch = (int)(gid & 15);
    const long long s = src[e];
    const long long d = dst[e];
    atomicAdd(&agg[d * NUM_CLS + cch], h[s * NUM_CLS + cch]);
}

// ---------------- finalize + graph readout sums ----------------
__global__ void gch_finalize(const float* __restrict__ agg,
                             const float* __restrict__ norm_dst,
                             const float* __restrict__ bias,
                             const long long* __restrict__ seg,
                             float* __restrict__ gsum,
                             float* __restrict__ gcnt, long long nWork) {
    long long gid = (long long)blockIdx.x * blockDim.x + threadIdx.x;
    if (gid >= nWork) return;
    const int i = (int)(gid >> 4);
    const int cch = (int)(gid & 15);
    const float v = agg[gid] * norm_dst[i] + bias[cch];
    const int g = (int)seg[i];
    atomicAdd(&gsum[g * NUM_CLS + cch], v);
    if (cch == 0) atomicAdd(&gcnt[g], 1.0f);
}

// ---------------- mean over nodes per graph ----------------
__global__ void gch_mean(const float* __restrict__ gsum,
                         const float* __restrict__ gcnt,
                         float* __restrict__ out, int n) {
    int i = blockIdx.x * blockDim.x + threadIdx.x;
    if (i >= n) return;
    out[i] = gsum[i] / fmaxf(gcnt[i >> 4], 1.0f);
}

extern "C" void kernel_launch(void* const* d_in, const int* in_sizes, int n_in,
                              void* d_out, int out_size, void* d_ws, size_t ws_size,
                              hipStream_t stream) {
    const float*     in_feat  = (const float*)d_in[0];
    const float*     W        = (const float*)d_in[1];
    const float*     b        = (const float*)d_in[2];
    const long long* edge_src = (const long long*)d_in[3];
    const long long* edge_dst = (const long long*)d_in[4];
    const long long* seg_ids  = (const long long*)d_in[5];
    float* out = (float*)d_out;

    // workspace layout (floats)
    float* ws       = (float*)d_ws;
    float* norm_src = ws;                                   // N  (deg -> rsqrt in place)
    float* norm_dst = norm_src + N_NODES;                   // N
    float* h        = norm_dst + N_NODES;                   // N*16
    float* agg      = h + (long long)N_NODES * NUM_CLS;     // N*16
    float* gsum     = agg + (long long)N_NODES * NUM_CLS;   // G*16
    float* gcnt     = gsum + NUM_GRAPHS * NUM_CLS;          // G

    // 1) zero accumulators (degrees, agg, gsum, gcnt). h is fully overwritten.
    {
        long long n0 = 2LL * N_NODES;
        gch_zero<<<(int)((n0 + 255) / 256), 256, 0, stream>>>(norm_src, n0);
        long long n1 = (long long)N_NODES * NUM_CLS + NUM_GRAPHS * NUM_CLS + NUM_GRAPHS;
        gch_zero<<<(int)((n1 + 255) / 256), 256, 0, stream>>>(agg, n1);
    }

    // 2) degrees
    gch_degrees<<<(N_EDGES + 255) / 256, 256, 0, stream>>>(edge_src, edge_dst,
                                                           norm_src, norm_dst, N_EDGES);
    // 3) norms in place
    gch_norms<<<(N_NODES + 255) / 256, 256, 0, stream>>>(norm_src, norm_dst, N_NODES);

    // 4) WMMA projection fused with src-norm scaling
    gch_project<<<(N_TILES + 7) / 8, 256, 0, stream>>>(in_feat, W, norm_src, h, N_TILES);

    // 5) edge aggregation (16 lanes per edge)
    {
        long long nWork = (long long)N_EDGES * NUM_CLS;     // 51.2M
        gch_aggregate<<<(int)((nWork + 255) / 256), 256, 0, stream>>>(edge_src, edge_dst,
                                                                      h, agg, nWork);
    }

    // 6) dst-norm + bias + segment sums
    {
        long long nWork = (long long)N_NODES * NUM_CLS;     // 1.6M
        gch_finalize<<<(int)((nWork + 255) / 256), 256, 0, stream>>>(agg, norm_dst, b,
                                                                     seg_ids, gsum, gcnt, nWork);
    }

    // 7) means -> output [256,16]
    gch_mean<<<(NUM_GRAPHS * NUM_CLS + 255) / 256, 256, 0, stream>>>(gsum, gcnt, out,
                                                                     NUM_GRAPHS * NUM_CLS);
}